// DivrocLoss_65987877535944
// MI455X (gfx1250) — compile-verified
//
#include <hip/hip_runtime.h>
#include <hip/hip_bf16.h>

#define GRIDSZ 256
#define NVOX (GRIDSZ * GRIDSZ * GRIDSZ)   // 16,777,216 voxels, 64 MB f32 (fits in 192MB L2)
#define RED_BLOCKS 512
#define RED_THREADS 256

typedef __attribute__((ext_vector_type(2))) float v2f;
typedef __attribute__((ext_vector_type(8))) float v8f;

// ---------------------------------------------------------------------------
// Kernel 1: zero the difference volume (float4 stores) and the scalar output.
// ---------------------------------------------------------------------------
__global__ void zero_kernel(float* __restrict__ vol, float* __restrict__ out) {
    const long tid    = (long)blockIdx.x * blockDim.x + threadIdx.x;
    const long stride = (long)gridDim.x * blockDim.x;
    float4* v4 = (float4*)vol;
    const long nv4 = NVOX / 4;
    for (long k = tid; k < nv4; k += stride)
        v4[k] = make_float4(0.f, 0.f, 0.f, 0.f);
    if (tid == 0) *out = 0.f;
}

// ---------------------------------------------------------------------------
// Kernel 2: fused trilinear scatter. Pred points splat +w, gt points splat -w
// into one difference volume (Huber needs the voxelwise difference, and the
// splat is linear, so one volume suffices). Atomics are non-returning relaxed
// agent-scope f32 adds -> single global_atomic_add_f32 instructions, all
// hitting the L2-resident volume.
// ---------------------------------------------------------------------------
__device__ __forceinline__ void splat_point(float* __restrict__ vol,
                                            float x, float y, float z,
                                            float sign) {
    // grid_sample align_corners=False unnormalization: ((p+1)*256 - 1)/2 = p*128 + 127.5
    const float ux = fmaf(x, 128.0f, 127.5f);
    const float uy = fmaf(y, 128.0f, 127.5f);
    const float uz = fmaf(z, 128.0f, 127.5f);
    const float x0f = floorf(ux), y0f = floorf(uy), z0f = floorf(uz);
    const float fx = ux - x0f, fy = uy - y0f, fz = uz - z0f;
    const int x0 = (int)x0f, y0 = (int)y0f, z0 = (int)z0f;
    const float wx[2] = {1.0f - fx, fx};
    const float wy[2] = {1.0f - fy, fy};
    const float wz[2] = {1.0f - fz, fz};
#pragma unroll
    for (int dz = 0; dz < 2; ++dz) {
        const int zi = z0 + dz;
        if ((unsigned)zi >= (unsigned)GRIDSZ) continue;
        const float sz = sign * wz[dz];
#pragma unroll
        for (int dy = 0; dy < 2; ++dy) {
            const int yi = y0 + dy;
            if ((unsigned)yi >= (unsigned)GRIDSZ) continue;
            const float szy = sz * wy[dy];
#pragma unroll
            for (int dx = 0; dx < 2; ++dx) {
                const int xi = x0 + dx;
                if ((unsigned)xi >= (unsigned)GRIDSZ) continue;
                const long idx = ((long)zi * GRIDSZ + yi) * GRIDSZ + xi;
                __hip_atomic_fetch_add(&vol[idx], szy * wx[dx],
                                       __ATOMIC_RELAXED, __HIP_MEMORY_SCOPE_AGENT);
            }
        }
    }
}

__global__ void splat_kernel(const float* __restrict__ pred,
                             const float* __restrict__ gt,
                             const float* __restrict__ coords,
                             float* __restrict__ vol, int n) {
    const int i = blockIdx.x * blockDim.x + threadIdx.x;
    if (i >= n) return;
    // Speculative prefetch ahead on the three streamed arrays
    // (gfx1250 global_prefetch_b8; OOB translation failures are dropped).
    __builtin_prefetch(coords + 3 * (i + 8192), 0, 0);
    __builtin_prefetch(pred   + 3 * (i + 8192), 0, 0);
    __builtin_prefetch(gt     + 3 * (i + 8192), 0, 0);

    const float cx = coords[3 * i + 0];
    const float cy = coords[3 * i + 1];
    const float cz = coords[3 * i + 2];
    const float px = pred[3 * i + 0];
    const float py = pred[3 * i + 1];
    const float pz = pred[3 * i + 2];
    const float gx = gt[3 * i + 0];
    const float gy = gt[3 * i + 1];
    const float gz = gt[3 * i + 2];

    splat_point(vol, cx + px, cy + py, cz + pz, +1.0f);
    splat_point(vol, cx + gx, cy + gy, cz + gz, -1.0f);
}

// ---------------------------------------------------------------------------
// Kernel 3: Huber reduce via V_WMMA_F32_16X16X4_F32.
// B = all-ones 4x16 => D[m][n] = sum_k A[m][k]; every A element reaches all 16
// columns, so sum(all C elements) = 16 * sum(all A elements) regardless of the
// A VGPR layout. Each wave folds 128 f32 Huber values per iteration through
// two independent WMMA accumulator chains (pipelines the D->C dependency) at
// full f32 precision. Trip count is a uniform SGPR loop so EXEC stays all-1s
// (required by WMMA) and no per-lane 64-bit compares are generated.
// ---------------------------------------------------------------------------
__device__ __forceinline__ float wave_allsum(float s) {
    // 5x ds_swizzle_b32 XOR-swap (group-of-32 mode: and=0x1f, xor in [14:10]).
    s += __int_as_float(__builtin_amdgcn_ds_swizzle(__float_as_int(s), 0x401F)); // SWAPX16
    s += __int_as_float(__builtin_amdgcn_ds_swizzle(__float_as_int(s), 0x201F)); // SWAPX8
    s += __int_as_float(__builtin_amdgcn_ds_swizzle(__float_as_int(s), 0x101F)); // SWAPX4
    s += __int_as_float(__builtin_amdgcn_ds_swizzle(__float_as_int(s), 0x081F)); // SWAPX2
    s += __int_as_float(__builtin_amdgcn_ds_swizzle(__float_as_int(s), 0x041F)); // SWAPX1
    return s;
}

__device__ __forceinline__ float huber(float d) {
    const float a = fabsf(d);
    return (a <= 1.0f) ? 0.5f * d * d : a - 0.5f;
}

__global__ void huber_reduce_kernel(const float* __restrict__ vol,
                                    float* __restrict__ out, int iters) {
    const int lane    = threadIdx.x & 31;
    const long wave   = ((long)blockIdx.x * blockDim.x + threadIdx.x) >> 5;
    const long nwaves = ((long)gridDim.x * blockDim.x) >> 5;

    v8f c0 = {};
    v8f c1 = {};
    v2f bones;
    bones[0] = 1.0f;
    bones[1] = 1.0f;

    // Each wave reads 32 float4 (128 voxels) per iteration, b128 per lane.
    const float4* p = (const float4*)vol + wave * 32 + lane;
    const long stride4 = nwaves * 32;       // grid stride in float4 units

#pragma unroll 4
    for (int it = 0; it < iters; ++it) {    // uniform scalar loop: EXEC untouched
        const float4 d = *p;
        v2f a01, a23;
        a01[0] = huber(d.x);
        a01[1] = huber(d.y);
        a23[0] = huber(d.z);
        a23[1] = huber(d.w);
        // 8 args: (neg_a, A, neg_b, B, c_mod, C, reuse_a, reuse_b)
        c0 = __builtin_amdgcn_wmma_f32_16x16x4_f32(false, a01, false, bones,
                                                   (short)0, c0, false, false);
        c1 = __builtin_amdgcn_wmma_f32_16x16x4_f32(false, a23, false, bones,
                                                   (short)0, c1, false, false);
        p += stride4;
    }

    float s = (c0[0] + c1[0]) + (c0[1] + c1[1]) + (c0[2] + c1[2]) + (c0[3] + c1[3])
            + (c0[4] + c1[4]) + (c0[5] + c1[5]) + (c0[6] + c1[6]) + (c0[7] + c1[7]);
    s = wave_allsum(s) * (1.0f / 16.0f);

    if (lane == 0)
        __hip_atomic_fetch_add(out, s, __ATOMIC_RELAXED, __HIP_MEMORY_SCOPE_AGENT);
}

// ---------------------------------------------------------------------------
// Launch: inputs are registration_pred [N,3] f32, registration_gt [N,3] f32,
// coords [N,3] f32; output is one f32 scalar; d_ws holds the 64 MB diff volume.
// ---------------------------------------------------------------------------
extern "C" void kernel_launch(void* const* d_in, const int* in_sizes, int n_in,
                              void* d_out, int out_size, void* d_ws, size_t ws_size,
                              hipStream_t stream) {
    const float* pred   = (const float*)d_in[0];
    const float* gt     = (const float*)d_in[1];
    const float* coords = (const float*)d_in[2];
    float* out = (float*)d_out;
    float* vol = (float*)d_ws;

    if (ws_size < (size_t)NVOX * sizeof(float)) return;  // need 64 MB scratch

    const int n = in_sizes[2] / 3;   // number of points

    zero_kernel<<<2048, 256, 0, stream>>>(vol, out);
    splat_kernel<<<(n + 255) / 256, 256, 0, stream>>>(pred, gt, coords, vol, n);

    // 512 blocks * 8 waves = 4096 waves; each iteration covers 4096*128 voxels.
    const long stride4 = (long)((RED_BLOCKS * RED_THREADS) / 32) * 32;  // float4/iter
    const int iters = (int)((NVOX / 4) / stride4);                      // = 32 exactly
    huber_reduce_kernel<<<RED_BLOCKS, RED_THREADS, 0, stream>>>(vol, out, iters);
}